// GoBERT_4020089389713
// MI455X (gfx1250) — compile-verified
//
#include <hip/hip_runtime.h>
#include <hip/hip_bf16.h>

typedef __attribute__((ext_vector_type(2))) float v2f;
typedef __attribute__((ext_vector_type(8))) float v8f;

#define NSEG 64

static __device__ __forceinline__ float eluf(float x) {
  return x > 0.f ? x : expm1f(x);
}

// float atomic max via monotone int mapping (works for mixed signs)
static __device__ __forceinline__ void atomicMaxF(float* addr, float val) {
  if (val >= 0.f) atomicMax((int*)addr, __float_as_int(val));
  else            atomicMin((unsigned int*)addr, __float_as_uint(val));
}

// ------------------------------ fill ------------------------------
__global__ void k_fill(float* __restrict__ p, float v, long n) {
  long i = (long)blockIdx.x * blockDim.x + threadIdx.x;
  long stride = (long)gridDim.x * blockDim.x;
  for (; i < n; i += stride) p[i] = v;
}

// --------- GEMM: Y[M,128] = X[M,128] @ W[128,128], fp32 WMMA ---------
// grid.x = ceil(M/16); block = 256 (8 waves, one 16x16 out tile each)
__global__ __launch_bounds__(256) void k_gemm128(const float* __restrict__ X,
                                                 const float* __restrict__ W,
                                                 float* __restrict__ Y, int M) {
  const int wave = threadIdx.x >> 5;
  const int lane = threadIdx.x & 31;
  const int half = lane >> 4;     // 0: K pair {0,1}, 1: K pair {2,3}
  const int idx  = lane & 15;
  const int m0 = blockIdx.x * 16;
  const int n0 = wave * 16;
  int row = m0 + idx; if (row >= M) row = M - 1;  // keep EXEC full for WMMA
  const float* __restrict__ xr = X + (long)row * 128;
  v8f c = {};
#pragma unroll
  for (int k0 = 0; k0 < 128; k0 += 4) {
    const int ka = k0 + 2 * half;
    v2f a, b;
    a.x = xr[ka];
    a.y = xr[ka + 1];
    b.x = W[ka * 128 + n0 + idx];
    b.y = W[(ka + 1) * 128 + n0 + idx];
    c = __builtin_amdgcn_wmma_f32_16x16x4_f32(false, a, false, b,
                                              (short)0, c, false, false);
  }
#pragma unroll
  for (int r = 0; r < 8; ++r) {
    const int orow = m0 + r + 8 * half;
    if (orow < M) Y[(long)orow * 128 + n0 + idx] = c[r];
  }
}

// ---- per-node attention logits: a_src[n,h], a_dst[n,h] (wave/node) ----
__global__ __launch_bounds__(256) void k_node_attn(const float* __restrict__ h,
    const float* __restrict__ att_s, const float* __restrict__ att_d,
    float* __restrict__ asrc, float* __restrict__ adst, int N) {
  const int n = (int)(((long)blockIdx.x * blockDim.x + threadIdx.x) >> 5);
  const int lane = threadIdx.x & 31;
  if (n >= N) return;
  const int c4 = lane * 4;         // element base; head = c4/32 = lane>>3
  const float4 hv = *(const float4*)(h + (long)n * 128 + c4);
  // att arrays are flat [H*C1]=[128] matching the row layout: index == c4+i
  float ps = hv.x * att_s[c4] + hv.y * att_s[c4 + 1] +
             hv.z * att_s[c4 + 2] + hv.w * att_s[c4 + 3];
  float pd = hv.x * att_d[c4] + hv.y * att_d[c4 + 1] +
             hv.z * att_d[c4 + 2] + hv.w * att_d[c4 + 3];
#pragma unroll
  for (int off = 1; off < 8; off <<= 1) {
    ps += __shfl_xor(ps, off, 32);
    pd += __shfl_xor(pd, off, 32);
  }
  if ((lane & 7) == 0) {
    const int head = lane >> 3;
    asrc[(long)n * 4 + head] = ps;
    adst[(long)n * 4 + head] = pd;
  }
}

// ---- edge pass 1: e = leakyrelu(a_src[src]+a_dst[dst]); seg-max; degree ----
__global__ void k_edge1(const int* __restrict__ ei,
                        const float* __restrict__ asrc,
                        const float* __restrict__ adst,
                        float* __restrict__ m, float* __restrict__ deg,
                        float* __restrict__ ebuf, int E, int N) {
  const long e = (long)blockIdx.x * blockDim.x + threadIdx.x;
  const long Ep = (long)E + N;
  if (e >= Ep) return;
  int s, d;
  if (e < E) { s = ei[e]; d = ei[(long)E + e]; } else { s = d = (int)(e - E); }
  const float4 av = *(const float4*)(asrc + (long)s * 4);
  const float4 bv = *(const float4*)(adst + (long)d * 4);
  float4 v;
  v.x = av.x + bv.x; v.y = av.y + bv.y; v.z = av.z + bv.z; v.w = av.w + bv.w;
  v.x = v.x > 0.f ? v.x : 0.2f * v.x;
  v.y = v.y > 0.f ? v.y : 0.2f * v.y;
  v.z = v.z > 0.f ? v.z : 0.2f * v.z;
  v.w = v.w > 0.f ? v.w : 0.2f * v.w;
  *(float4*)(ebuf + e * 4) = v;
  float* md = m + (long)d * 4;
  atomicMaxF(md + 0, v.x); atomicMaxF(md + 1, v.y);
  atomicMaxF(md + 2, v.z); atomicMaxF(md + 3, v.w);
  atomicAdd(deg + d, 1.0f);
}

// ---- edge pass 2: exp(e - m[dst]); seg-sum; ebuf <- exp ----
__global__ void k_edge2(const int* __restrict__ ei,
                        float* __restrict__ ebuf,
                        const float* __restrict__ m, float* __restrict__ ssum,
                        int E, int N) {
  const long e = (long)blockIdx.x * blockDim.x + threadIdx.x;
  const long Ep = (long)E + N;
  if (e >= Ep) return;
  int d;
  if (e < E) d = ei[(long)E + e]; else d = (int)(e - E);
  float4 v = *(const float4*)(ebuf + e * 4);
  float4 mm = *(const float4*)(m + (long)d * 4);
  if (mm.x < -3.0e38f) mm.x = 0.f;
  if (mm.y < -3.0e38f) mm.y = 0.f;
  if (mm.z < -3.0e38f) mm.z = 0.f;
  if (mm.w < -3.0e38f) mm.w = 0.f;
  float4 ex;
  ex.x = expf(v.x - mm.x); ex.y = expf(v.y - mm.y);
  ex.z = expf(v.z - mm.z); ex.w = expf(v.w - mm.w);
  float* sd = ssum + (long)d * 4;
  atomicAdd(sd + 0, ex.x); atomicAdd(sd + 1, ex.y);
  atomicAdd(sd + 2, ex.z); atomicAdd(sd + 3, ex.w);
  *(float4*)(ebuf + e * 4) = ex;
}

// ---- edge pass 3: out1[dst] += h[src] * alpha (wave/edge, 4 floats/lane) ----
__global__ __launch_bounds__(256) void k_gat_scatter(const int* __restrict__ ei,
    const float* __restrict__ h, const float* __restrict__ ebuf,
    const float* __restrict__ ssum, float* __restrict__ out1, int E, int N) {
  const long wid = ((long)blockIdx.x * blockDim.x + threadIdx.x) >> 5;
  const int lane = threadIdx.x & 31;
  const long Ep = (long)E + N;
  if (wid >= Ep) return;
  int s, d;
  if (wid < E) { s = ei[wid]; d = ei[(long)E + wid]; } else { s = d = (int)(wid - E); }
  const int head = lane >> 3;
  const float alpha = ebuf[wid * 4 + head] / (ssum[(long)d * 4 + head] + 1e-16f);
  const float4 hv = *(const float4*)(h + (long)s * 128 + lane * 4);
  float* o = out1 + (long)d * 128 + lane * 4;
  atomicAdd(o + 0, hv.x * alpha);
  atomicAdd(o + 1, hv.y * alpha);
  atomicAdd(o + 2, hv.z * alpha);
  atomicAdd(o + 3, hv.w * alpha);
}

// ---- out1 epilogue: elu(acc + b_gat) * bn1_w + bn1_b (in place) ----
__global__ void k_out1_fin(float* __restrict__ out1, const float* __restrict__ b,
                           const float* __restrict__ w, const float* __restrict__ bb,
                           long n) {
  const long i = (long)blockIdx.x * blockDim.x + threadIdx.x;
  if (i >= n) return;
  const int c = (int)(i & 127);
  float v = out1[i] + b[c];
  v = eluf(v);
  out1[i] = v * w[c] + bb[c];
}

// ---- GCN scatter: out2[dst] += h2[src] * rsqrt-norm (wave/edge) ----
__global__ __launch_bounds__(256) void k_gcn_scatter(const int* __restrict__ ei,
    const float* __restrict__ h2, const float* __restrict__ deg,
    float* __restrict__ out2, int E, int N) {
  const long wid = ((long)blockIdx.x * blockDim.x + threadIdx.x) >> 5;
  const int lane = threadIdx.x & 31;
  const long Ep = (long)E + N;
  if (wid >= Ep) return;
  int s, d;
  if (wid < E) { s = ei[wid]; d = ei[(long)E + wid]; } else { s = d = (int)(wid - E); }
  const float ds = deg[s] < 1.f ? 1.f : deg[s];
  const float dd = deg[d] < 1.f ? 1.f : deg[d];
  const float norm = rsqrtf(ds) * rsqrtf(dd);
  const float4 hv = *(const float4*)(h2 + (long)s * 128 + lane * 4);
  float* o = out2 + (long)d * 128 + lane * 4;
  atomicAdd(o + 0, hv.x * norm);
  atomicAdd(o + 1, hv.y * norm);
  atomicAdd(o + 2, hv.z * norm);
  atomicAdd(o + 3, hv.w * norm);
}

// ---- out2 epilogue + gate logit + per-graph max (wave/node) ----
__global__ __launch_bounds__(256) void k_out2_gate(float* __restrict__ out2,
    const float* __restrict__ bgcn, const float* __restrict__ bn2w,
    const float* __restrict__ bn2b, const float* __restrict__ Wgate,
    const float* __restrict__ bgate, const int* __restrict__ batch,
    float* __restrict__ gate, float* __restrict__ gm, int N) {
  const int n = (int)(((long)blockIdx.x * blockDim.x + threadIdx.x) >> 5);
  const int lane = threadIdx.x & 31;
  if (n >= N) return;
  const int c = lane * 4;
  float* row = out2 + (long)n * 128 + c;
  float4 v = *(const float4*)row;
  v.x = eluf(v.x + bgcn[c]) * bn2w[c] + bn2b[c];
  v.y = eluf(v.y + bgcn[c + 1]) * bn2w[c + 1] + bn2b[c + 1];
  v.z = eluf(v.z + bgcn[c + 2]) * bn2w[c + 2] + bn2b[c + 2];
  v.w = eluf(v.w + bgcn[c + 3]) * bn2w[c + 3] + bn2b[c + 3];
  *(float4*)row = v;
  float g = v.x * Wgate[c] + v.y * Wgate[c + 1] +
            v.z * Wgate[c + 2] + v.w * Wgate[c + 3];
#pragma unroll
  for (int off = 1; off < 32; off <<= 1) g += __shfl_xor(g, off, 32);
  if (lane == 0) {
    g += bgate[0];
    gate[n] = g;
    atomicMaxF(gm + batch[n], g);
  }
}

// ---- gate softmax numerator + per-graph sum (thread/node) ----
__global__ void k_gate2(const float* __restrict__ gate, const float* __restrict__ gm,
                        float* __restrict__ gs, float* __restrict__ gexp,
                        const int* __restrict__ batch, int N) {
  const int n = (int)((long)blockIdx.x * blockDim.x + threadIdx.x);
  if (n >= N) return;
  const int b = batch[n];
  float mm = gm[b];
  if (mm < -3.0e38f) mm = 0.f;
  const float ge = expf(gate[n] - mm);
  atomicAdd(gs + b, ge);
  gexp[n] = ge;
}

// ---- pooling: graph_rep[b] += w * out2[n] (wave/node) ----
__global__ __launch_bounds__(256) void k_pool(const float* __restrict__ out2,
    const float* __restrict__ gexp, const float* __restrict__ gs,
    const int* __restrict__ batch, float* __restrict__ grep, int N) {
  const int n = (int)(((long)blockIdx.x * blockDim.x + threadIdx.x) >> 5);
  const int lane = threadIdx.x & 31;
  if (n >= N) return;
  const int b = batch[n];
  const float w = gexp[n] / (gs[b] + 1e-16f);
  const int c = lane * 4;
  const float4 v = *(const float4*)(out2 + (long)n * 128 + c);
  float* o = grep + (long)b * 128 + c;
  atomicAdd(o + 0, v.x * w);
  atomicAdd(o + 1, v.y * w);
  atomicAdd(o + 2, v.z * w);
  atomicAdd(o + 3, v.w * w);
}

// ---- final: out[g] = grep[g] . W_fc + b_fc (one wave per graph) ----
__global__ void k_final(const float* __restrict__ grep, const float* __restrict__ Wfc,
                        const float* __restrict__ bfc, float* __restrict__ out) {
  const int g = blockIdx.x;
  const int lane = threadIdx.x;
  const int c = lane * 4;
  const float4 v = *(const float4*)(grep + (long)g * 128 + c);
  float p = v.x * Wfc[c] + v.y * Wfc[c + 1] + v.z * Wfc[c + 2] + v.w * Wfc[c + 3];
#pragma unroll
  for (int off = 1; off < 32; off <<= 1) p += __shfl_xor(p, off, 32);
  if (lane == 0) out[g] = p + bfc[0];
}

extern "C" void kernel_launch(void* const* d_in, const int* in_sizes, int n_in,
                              void* d_out, int out_size, void* d_ws, size_t ws_size,
                              hipStream_t stream) {
  const float* x      = (const float*)d_in[0];
  const int*   ei     = (const int*)d_in[1];
  const int*   batch  = (const int*)d_in[2];
  const float* Wgat   = (const float*)d_in[3];
  const float* attS   = (const float*)d_in[4];
  const float* attD   = (const float*)d_in[5];
  const float* bgat   = (const float*)d_in[6];
  const float* bn1w   = (const float*)d_in[7];
  const float* bn1b   = (const float*)d_in[8];
  const float* Wgcn   = (const float*)d_in[9];
  const float* bgcn   = (const float*)d_in[10];
  const float* bn2w   = (const float*)d_in[11];
  const float* bn2b   = (const float*)d_in[12];
  const float* Wgate  = (const float*)d_in[13];
  const float* bgate  = (const float*)d_in[14];
  const float* Wfc    = (const float*)d_in[15];
  const float* bfc    = (const float*)d_in[16];

  const int N  = in_sizes[2];          // batch[] has N entries
  const int E  = in_sizes[1] / 2;
  const long Ep = (long)E + N;
  const long NF = (long)N * 128;

  float* ws = (float*)d_ws;
  float* A    = ws;                 // h, then h2
  float* B    = ws + NF;            // out1 acc/final, then out2 acc/final
  float* asrc = ws + 2 * NF;
  float* adst = asrc + (long)N * 4;
  float* m    = adst + (long)N * 4;
  float* ssum = m    + (long)N * 4;
  float* deg  = ssum + (long)N * 4;
  float* gate = deg  + N;
  float* gexp = gate + N;
  float* gm   = gexp + N;
  float* gs   = gm + NSEG;
  float* grep = gs + NSEG;
  float* ebuf = grep + (long)NSEG * 128;  // Ep*4 floats

  const float NEG_INF = -__builtin_inff();
  auto fgrid = [](long n) { long g = (n + 255) / 256; return (int)(g > 4096 ? 4096 : g); };

  // init accumulators (workspace is not re-poisoned between replays)
  k_fill<<<fgrid(NF), 256, 0, stream>>>(B, 0.f, NF);
  k_fill<<<fgrid((long)N * 4), 256, 0, stream>>>(m, NEG_INF, (long)N * 4);
  k_fill<<<fgrid((long)N * 4), 256, 0, stream>>>(ssum, 0.f, (long)N * 4);
  k_fill<<<fgrid(N), 256, 0, stream>>>(deg, 0.f, N);
  k_fill<<<1, 64, 0, stream>>>(gm, NEG_INF, NSEG);
  k_fill<<<1, 64, 0, stream>>>(gs, 0.f, NSEG);
  k_fill<<<32, 256, 0, stream>>>(grep, 0.f, (long)NSEG * 128);

  const int gemmGrid = (N + 15) / 16;
  const int waveNodeGrid = (N + 7) / 8;            // wave per node, 8 waves/block
  const int edgeGrid = (int)((Ep + 255) / 256);    // thread per edge
  const int waveEdgeGrid = (int)((Ep + 7) / 8);    // wave per edge

  // GAT
  k_gemm128<<<gemmGrid, 256, 0, stream>>>(x, Wgat, A, N);                 // h
  k_node_attn<<<waveNodeGrid, 256, 0, stream>>>(A, attS, attD, asrc, adst, N);
  k_edge1<<<edgeGrid, 256, 0, stream>>>(ei, asrc, adst, m, deg, ebuf, E, N);
  k_edge2<<<edgeGrid, 256, 0, stream>>>(ei, ebuf, m, ssum, E, N);
  k_gat_scatter<<<waveEdgeGrid, 256, 0, stream>>>(ei, A, ebuf, ssum, B, E, N);
  k_out1_fin<<<(int)((NF + 255) / 256), 256, 0, stream>>>(B, bgat, bn1w, bn1b, NF);

  // GCN
  k_gemm128<<<gemmGrid, 256, 0, stream>>>(B, Wgcn, A, N);                 // h2 (h dead)
  k_fill<<<fgrid(NF), 256, 0, stream>>>(B, 0.f, NF);                      // out2 acc
  k_gcn_scatter<<<waveEdgeGrid, 256, 0, stream>>>(ei, A, deg, B, E, N);

  // Global attention pooling
  k_out2_gate<<<waveNodeGrid, 256, 0, stream>>>(B, bgcn, bn2w, bn2b, Wgate, bgate,
                                                batch, gate, gm, N);
  k_gate2<<<(N + 255) / 256, 256, 0, stream>>>(gate, gm, gs, gexp, batch, N);
  k_pool<<<waveNodeGrid, 256, 0, stream>>>(B, gexp, gs, batch, grep, N);
  k_final<<<NSEG, 32, 0, stream>>>(grep, Wfc, bfc, (float*)d_out);
}